// GaussianNLLLoss_19155554140741
// MI455X (gfx1250) — compile-verified
//
#include <hip/hip_runtime.h>
#include <math.h>

typedef __attribute__((ext_vector_type(2))) float v2f;
typedef __attribute__((ext_vector_type(8))) float v8f;

#define BLOCK 256
#define GRID  2048
#define WAVES_PER_BLOCK (BLOCK / 32)

// Per-element log-likelihood with scalar constants pre-folded:
//   loglik = K + log(0.5 + 0.5*erf(c * resid)) - resid^2 * inv2s2
__device__ __forceinline__ float loglik_elem(float yp, float yt,
                                             float K, float c, float inv2s2) {
    float resid = yt - yp;
    float t     = c * resid;
    float ff    = fmaf(0.5f, erff(t), 0.5f);
    return K + logf(ff) - resid * resid * inv2s2;
}

__global__ __launch_bounds__(BLOCK)
void nll_partial_kernel(const float* __restrict__ y_pred,
                        const float* __restrict__ y_true,
                        const float* __restrict__ log_std_v,
                        const float* __restrict__ log_std_u,
                        float* __restrict__ partial,
                        long long n) {
    // Scalar precompute (broadcast loads).
    const float std_v  = expf(log_std_v[0]);
    const float std_u  = expf(log_std_u[0]);
    const float sigma2 = std_v * std_v + std_u * std_u;
    const float sigma  = sqrtf(sigma2);
    const float K      = -0.5f * logf(1.57079632679489661923f)  // log(pi/2)
                         - 0.5f * logf(sigma2);
    const float c      = -(std_u / std_v) / sigma * 0.70710678118654752440f;
    const float inv2s2 = 0.5f / sigma2;

    const long long V = n >> 2;                 // float4 count
    const int lane = threadIdx.x & 31;
    // Force wave-uniformity into an SGPR so the loop is scalar-controlled:
    // EXEC is provably all-ones at every WMMA.
    const int waveInBlk = __builtin_amdgcn_readfirstlane((int)(threadIdx.x >> 5));
    const long long gwave  = (long long)blockIdx.x * WAVES_PER_BLOCK + waveInBlk;
    const long long nwaves = (long long)gridDim.x * WAVES_PER_BLOCK;
    const long long fullChunks = V >> 5;        // chunks of 32 float4 = 128 floats

    const float4* __restrict__ yp4 = (const float4*)y_pred;
    const float4* __restrict__ yt4 = (const float4*)y_true;

    // WMMA accumulator: D += A x ones. Columns of D are identical; fp32 exact.
    v8f acc = {0.f, 0.f, 0.f, 0.f, 0.f, 0.f, 0.f, 0.f};
    const v2f bones = {1.0f, 1.0f};

    // ---- Main loop: full chunks only, no clamps, no masks, scalar branch ----
    for (long long ch = gwave; ch < fullChunks; ch += nwaves) {
        long long vi = (ch << 5) + lane;
        float4 p = yp4[vi];
        float4 t = yt4[vi];

        // Prefetch the chunk this wave touches next iteration (clamped, scalar math).
        long long pch = ch + nwaves;
        if (pch >= fullChunks) pch = ch;        // uniform condition -> s_cmp
        long long pvi = (pch << 5) + lane;
        __builtin_prefetch(&yp4[pvi], 0, 3);
        __builtin_prefetch(&yt4[pvi], 0, 3);

        float l0 = loglik_elem(p.x, t.x, K, c, inv2s2);
        float l1 = loglik_elem(p.y, t.y, K, c, inv2s2);
        float l2 = loglik_elem(p.z, t.z, K, c, inv2s2);
        float l3 = loglik_elem(p.w, t.w, K, c, inv2s2);

        v2f a01 = {l0, l1};
        v2f a23 = {l2, l3};
        acc = __builtin_amdgcn_wmma_f32_16x16x4_f32(false, a01, false, bones,
                                                    (short)0, acc, false, false);
        acc = __builtin_amdgcn_wmma_f32_16x16x4_f32(false, a23, false, bones,
                                                    (short)0, acc, false, false);
    }

    // ---- Partial last chunk (V % 32 float4s): one wave, scalar accumulate ----
    float extra = 0.f;
    if ((V & 31) && gwave == (fullChunks % nwaves)) {
        long long vi = (fullChunks << 5) + lane;
        if (vi < V) {
            float4 p = yp4[vi];
            float4 t = yt4[vi];
            extra = loglik_elem(p.x, t.x, K, c, inv2s2)
                  + loglik_elem(p.y, t.y, K, c, inv2s2)
                  + loglik_elem(p.z, t.z, K, c, inv2s2)
                  + loglik_elem(p.w, t.w, K, c, inv2s2);
        }
    }
    // Reduce `extra` across the wave (all lanes end with the wave total).
    #pragma unroll
    for (int off = 16; off > 0; off >>= 1)
        extra += __shfl_xor(extra, off, 32);

    // Lane-local: sum this lane's 8 row entries of D.
    float s = acc[0] + acc[1] + acc[2] + acc[3] + acc[4] + acc[5] + acc[6] + acc[7];
    // Columns identical within each 16-lane half: wave total = lane0 + lane16.
    float wsum = __shfl(s, 0, 32) + __shfl(s, 16, 32) + extra;

    __shared__ float lsum[WAVES_PER_BLOCK];
    if (lane == 0) lsum[waveInBlk] = wsum;
    __syncthreads();

    if (threadIdx.x == 0) {
        float b = 0.f;
        #pragma unroll
        for (int i = 0; i < WAVES_PER_BLOCK; ++i) b += lsum[i];
        // Scalar tail (n % 4), handled once by block 0 only.
        if (blockIdx.x == 0) {
            for (long long i = (V << 2); i < n; ++i)
                b += loglik_elem(y_pred[i], y_true[i], K, c, inv2s2);
        }
        partial[blockIdx.x] = b;
    }
}

__global__ __launch_bounds__(256)
void nll_finalize_kernel(const float* __restrict__ partial,
                         float* __restrict__ out,
                         long long n, int nparts) {
    __shared__ float red[256];
    float s = 0.f;
    for (int i = threadIdx.x; i < nparts; i += 256) s += partial[i];
    red[threadIdx.x] = s;
    __syncthreads();
    for (int off = 128; off > 0; off >>= 1) {
        if (threadIdx.x < off) red[threadIdx.x] += red[threadIdx.x + off];
        __syncthreads();
    }
    if (threadIdx.x == 0) out[0] = -red[0] / (float)n;
}

extern "C" void kernel_launch(void* const* d_in, const int* in_sizes, int n_in,
                              void* d_out, int out_size, void* d_ws, size_t ws_size,
                              hipStream_t stream) {
    const float* y_pred    = (const float*)d_in[0];
    const float* y_true    = (const float*)d_in[1];
    const float* log_std_v = (const float*)d_in[2];
    const float* log_std_u = (const float*)d_in[3];
    float* out = (float*)d_out;
    float* ws  = (float*)d_ws;           // GRID floats of scratch (8 KiB)
    long long n = (long long)in_sizes[0];

    nll_partial_kernel<<<GRID, BLOCK, 0, stream>>>(y_pred, y_true,
                                                   log_std_v, log_std_u, ws, n);
    nll_finalize_kernel<<<1, 256, 0, stream>>>(ws, out, n, GRID);
}